// OctreeConv_77936476553757
// MI455X (gfx1250) — compile-verified
//
#include <hip/hip_runtime.h>
#include <hip/hip_bf16.h>

typedef float v2f __attribute__((ext_vector_type(2)));
typedef float v8f __attribute__((ext_vector_type(8)));

#define NPTS   300000
#define CIN    32
#define COUT   64
#define KTAPS  27
#define NTILES (NPTS / 16)        // 18750, exact
#define TAPCHUNK 7                // taps staged in LDS per pass
#define NCHUNK  ((KTAPS + TAPCHUNK - 1) / TAPCHUNK)   // 4
#define WAVES_PER_BLOCK 8

// LDS layout per chunk: for tap t, cin-chunk j (cin = 4j..4j+3), cout n:
//   sw[(t*8 + j)*256 + n*4 + kk] = w[k0+t][4j+kk][n]
// so lane L reads its B fragment pair (K = 2*laneHi, 2*laneHi+1) as one b64.
__global__ __launch_bounds__(256) void octree_conv_wmma(
    const float* __restrict__ x,      // [N, CIN]
    const float* __restrict__ w,      // [K, CIN, COUT]
    const int*   __restrict__ neigh,  // [N, K]
    float*       __restrict__ out)    // [N, COUT]
{
    __shared__ float sw[TAPCHUNK * 8 * 256];   // 57,344 B

    const int tid    = threadIdx.x;
    const int lane   = tid & 31;
    const int wave   = tid >> 5;
    const int row16  = lane & 15;      // M (and B-col N) index
    const int laneHi = lane >> 4;      // selects K-pair {0,1} vs {2,3}
    const int tile   = blockIdx.x * WAVES_PER_BLOCK + wave;
    const bool live  = (tile < NTILES);

    v8f acc0 = {}, acc1 = {}, acc2 = {}, acc3 = {};

    const int* np = neigh + (size_t)(tile < NTILES ? tile : 0) * 16 * KTAPS
                          + (size_t)row16 * KTAPS;

    for (int chunk = 0; chunk < NCHUNK; ++chunk) {
        const int k0   = chunk * TAPCHUNK;
        const int ntap = (KTAPS - k0) < TAPCHUNK ? (KTAPS - k0) : TAPCHUNK;

        // ---- cooperative transposed weight stage into LDS ----
        const int total = ntap * CIN * COUT;   // ntap*2048
        for (int s = tid; s < total; s += 256) {
            const int t  = s >> 11;            // / (CIN*COUT)
            const int r  = s & 2047;
            const int c  = r >> 6;             // cin 0..31
            const int n  = r & 63;             // cout
            const float v = w[(size_t)((k0 + t) * CIN + c) * COUT + n];
            const int j  = c >> 2;
            const int kk = c & 3;
            sw[(t * 8 + j) * 256 + n * 4 + kk] = v;
        }
        __syncthreads();

        // ---- compute: gather A, ds-load B, WMMA ----
        if (live) {
            for (int t = 0; t < ntap; ++t) {
                int idx = np[k0 + t];
                const float m = (idx >= 0) ? 1.0f : 0.0f;
                idx = idx < 0 ? 0 : idx;
                const float* rp = x + (size_t)idx * CIN + 2 * laneHi;
                const float* sb0 = sw + t * 2048 + row16 * 4 + laneHi * 2;
#pragma unroll
                for (int j = 0; j < 8; ++j) {
                    v2f a = *(const v2f*)(rp + 4 * j);   // channels 4j+2*laneHi, +1
                    a = a * m;                            // mask invalid neighbor
                    const float* sb = sb0 + j * 256;
                    v2f b0 = *(const v2f*)(sb);          // cout block 0
                    v2f b1 = *(const v2f*)(sb + 64);     // cout block 1
                    v2f b2 = *(const v2f*)(sb + 128);    // cout block 2
                    v2f b3 = *(const v2f*)(sb + 192);    // cout block 3
                    acc0 = __builtin_amdgcn_wmma_f32_16x16x4_f32(
                        false, a, false, b0, (short)0, acc0, false, false);
                    acc1 = __builtin_amdgcn_wmma_f32_16x16x4_f32(
                        false, a, false, b1, (short)0, acc1, false, false);
                    acc2 = __builtin_amdgcn_wmma_f32_16x16x4_f32(
                        false, a, false, b2, (short)0, acc2, false, false);
                    acc3 = __builtin_amdgcn_wmma_f32_16x16x4_f32(
                        false, a, false, b3, (short)0, acc3, false, false);
                }
            }
        }
        __syncthreads();   // before next chunk overwrites LDS
    }

    // ---- store D: VGPR v holds M = v + 8*laneHi, N = row16 (per 16x16 block) ----
    if (live) {
        const int mbase = tile * 16 + 8 * laneHi;
#pragma unroll
        for (int v = 0; v < 8; ++v) {
            float* op = out + (size_t)(mbase + v) * COUT + row16;
            op[0]  = acc0[v];
            op[16] = acc1[v];
            op[32] = acc2[v];
            op[48] = acc3[v];
        }
    }
}

extern "C" void kernel_launch(void* const* d_in, const int* in_sizes, int n_in,
                              void* d_out, int out_size, void* d_ws, size_t ws_size,
                              hipStream_t stream) {
    (void)in_sizes; (void)n_in; (void)d_ws; (void)ws_size; (void)out_size;
    const float* x     = (const float*)d_in[0];
    const float* w     = (const float*)d_in[1];
    const int*   neigh = (const int*)d_in[2];
    float*       out   = (float*)d_out;

    const int grid = (NTILES + WAVES_PER_BLOCK - 1) / WAVES_PER_BLOCK;  // 2344
    octree_conv_wmma<<<grid, 256, 0, stream>>>(x, w, neigh, out);
}